// ProjectionLoss_1580547967532
// MI455X (gfx1250) — compile-verified
//
#include <hip/hip_runtime.h>
#include <math.h>

// ---------------------------------------------------------------------------
// ProjectionLoss for MI455X (gfx1250, wave32)
//   dist(p,g) = ||p||^2 - 2 p.g + ||g||^2
//   -2 p.g  -> V_WMMA_F32_16X16X4_F32 (K=3 padded to 4)
//   +||g||^2 folded into the WMMA C operand (per-column broadcast)
//   ||p||^2 is row-constant -> added after top-k selection (order-invariant)
// Block = 256 threads = 8 waves. All 8 waves share the same 32 pred rows but
// scan disjoint N-segments (1024 cols each) -> 2048 waves of parallelism.
// Per-wave top-8 lists merged through LDS, epilogue + atomic reduction.
// ---------------------------------------------------------------------------

#define KNN 8
constexpr int Bb = 4, Mm = 2048, Nn = 8192;
constexpr int WAVES = 8;              // waves per block
constexpr int ROWS  = 32;             // pred rows per block (2 WMMA tiles)
constexpr int SEG   = Nn / WAVES;     // 1024 columns per wave
constexpr float INV_SP2 = 1.0f / (0.03f * 0.03f);            // 1/sigma_p^2
constexpr float INV_ANG = 1.0f / 0.03407417371093169f;       // 1/(1-cos(sigma_n))

typedef float v2f __attribute__((ext_vector_type(2)));
typedef float v8f __attribute__((ext_vector_type(8)));

__global__ void projloss_zero(float* out) { out[0] = 0.0f; }

__global__ __launch_bounds__(256) void projloss_kernel(
    const float* __restrict__ preds,
    const float* __restrict__ gts,
    const float* __restrict__ normals,
    float* __restrict__ out)
{
  __shared__ float s_tile[WAVES * ROWS * 16];   // 16 KB: per-wave 32x16 dist tiles
  __shared__ float s_md[WAVES * ROWS * KNN];    // 8 KB: partial top-8 dists
  __shared__ int   s_mi[WAVES * ROWS * KNN];    // 8 KB: partial top-8 indices

  const int tid  = threadIdx.x;
  const int lane = tid & 31;
  const int w    = tid >> 5;
  const int blk  = blockIdx.x;            // 256 blocks
  const int b    = blk >> 6;              // 64 blocks per batch (2048/32)
  const int m_base = (blk & 63) * ROWS;

  const float* __restrict__ Pb = preds   + (size_t)b * Mm * 3;
  const float* __restrict__ Gb = gts     + (size_t)b * Nn * 3;
  const float* __restrict__ Nb = normals + (size_t)b * Nn * 3;

  // ---- A fragments: 16x4 fp32, lanes 0-15 hold K={0,1}, lanes 16-31 K={2,3}
  const int mlo = lane & 15;
  const int kb  = (lane >> 4) * 2;        // 0 or 2
  v2f a0, a1;
  {
    const float* P0 = Pb + (size_t)(m_base + mlo) * 3;
    const float* P1 = Pb + (size_t)(m_base + mlo + 16) * 3;
    a0.x = (kb == 0) ? P0[0] : P0[2];
    a0.y = (kb == 0) ? P0[1] : 0.0f;       // K=3 pad
    a1.x = (kb == 0) ? P1[0] : P1[2];
    a1.y = (kb == 0) ? P1[1] : 0.0f;
  }

  // ---- register-resident sorted top-8 (ascending) ----
  float bd[KNN]; int bi[KNN];
#pragma unroll
  for (int k = 0; k < KNN; ++k) { bd[k] = 3.0e38f; bi[k] = 0; }

  const int n_start  = w * SEG;
  const int n_end    = n_start + SEG;
  const int lds_base = w * ROWS * 16;
  const int col      = lane & 15;
  const int rbase    = (lane >> 4) * 8;

  for (int n0 = n_start; n0 < n_end; n0 += 16) {
    const int n = n0 + col;
    const float* G = Gb + (size_t)n * 3;
    if (n0 + 64 < n_end)
      __builtin_prefetch(Gb + (size_t)(n + 64) * 3, 0, 1);   // global_prefetch_b8

    const float gx = G[0], gy = G[1], gz = G[2];
    const float gsq = gx * gx + gy * gy + gz * gz;

    // B fragment: 4x16 fp32, column = lane&15; values = -2*g (K=3 pad 0)
    v2f bf;
    bf.x = (kb == 0) ? -2.0f * gx : -2.0f * gz;
    bf.y = (kb == 0) ? -2.0f * gy : 0.0f;

    // C operand: broadcast ||g_n||^2 into all 8 accumulators of this column
    v8f c;
#pragma unroll
    for (int v = 0; v < 8; ++v) c[v] = gsq;

    // D = A*B + C  ->  -2 p.g + ||g||^2   (16x16 tile each)
    v8f d0 = __builtin_amdgcn_wmma_f32_16x16x4_f32(
        false, a0, false, bf, (short)0, c, false, false);
    v8f d1 = __builtin_amdgcn_wmma_f32_16x16x4_f32(
        false, a1, false, bf, (short)0, c, false, false);

    // scatter D tiles into per-wave LDS as row-major 32x16
#pragma unroll
    for (int v = 0; v < 8; ++v) {
      s_tile[lds_base + (rbase + v) * 16 + col]      = d0[v];
      s_tile[lds_base + (16 + rbase + v) * 16 + col] = d1[v];
    }
    // wave-private region: only need DS counter drained, no barrier
    asm volatile("s_wait_dscnt 0" ::: "memory");

    // each lane owns pred row = lane; stream its 16 candidates back (b128)
    const float4* rowp = (const float4*)&s_tile[lds_base + lane * 16];
#pragma unroll
    for (int q = 0; q < 4; ++q) {
      float4 v4 = rowp[q];
      float dv[4] = {v4.x, v4.y, v4.z, v4.w};
#pragma unroll
      for (int t = 0; t < 4; ++t) {
        float nd = dv[t];
        int   ni = n0 + q * 4 + t;
        if (nd <= bd[KNN - 1]) {            // cheap reject vs current worst
#pragma unroll
          for (int k = 0; k < KNN; ++k) {   // sorted bubble-insert
            const bool le = nd <= bd[k];
            const float td = le ? bd[k] : nd;
            bd[k] = le ? nd : bd[k];
            nd = td;
            const int ti = le ? bi[k] : ni;
            bi[k] = le ? ni : bi[k];
            ni = ti;
          }
        }
      }
    }
  }

  // ---- publish per-wave partial top-8, merge across the 8 segments ----
#pragma unroll
  for (int k = 0; k < KNN; ++k) {
    s_md[(w * ROWS + lane) * KNN + k] = bd[k];
    s_mi[(w * ROWS + lane) * KNN + k] = bi[k];
  }
  __syncthreads();

  if (w == 0) {
    float fd[KNN]; int fi[KNN];
#pragma unroll
    for (int k = 0; k < KNN; ++k) { fd[k] = 3.0e38f; fi[k] = 0; }

    for (int s = 0; s < WAVES; ++s) {
#pragma unroll
      for (int k = 0; k < KNN; ++k) {
        float nd = s_md[(s * ROWS + lane) * KNN + k];
        int   ni = s_mi[(s * ROWS + lane) * KNN + k];
        if (nd <= fd[KNN - 1]) {
#pragma unroll
          for (int k2 = 0; k2 < KNN; ++k2) {
            const bool le = nd <= fd[k2];
            const float td = le ? fd[k2] : nd;
            fd[k2] = le ? nd : fd[k2];
            nd = td;
            const int ti = le ? fi[k2] : ni;
            fi[k2] = le ? ni : fi[k2];
            ni = ti;
          }
        }
      }
    }

    // ---- epilogue: weighted projection displacement for this pred point ----
    const int m = m_base + lane;
    const float* P = Pb + (size_t)m * 3;
    const float px = P[0], py = P[1], pz = P[2];
    const float psq = px * px + py * py + pz * pz;

    const float* E = Nb + (size_t)fi[0] * 3;   // nearest neighbor's normal
    const float e0 = E[0], e1 = E[1], e2 = E[2];

    float sw = 0.0f, swp = 0.0f;
#pragma unroll
    for (int k = 0; k < KNN; ++k) {
      const int i = fi[k];
      const float* G  = Gb + (size_t)i * 3;
      const float* NN = Nb + (size_t)i * 3;
      const float gx = G[0], gy = G[1], gz = G[2];
      const float nx = NN[0], ny = NN[1], nz = NN[2];
      const float d  = fmaxf(fd[k] + psq, 0.0f);     // restore ||p||^2 term
      const float dw = __expf(-d * INV_SP2);
      const float inner = nx * e0 + ny * e1 + nz * e2;
      const float aw = __expf(-(1.0f - inner) * INV_ANG);
      const float wgt = dw * aw;
      const float proj = fabsf((px - gx) * nx + (py - gy) * ny + (pz - gz) * nz);
      sw  += wgt;
      swp += proj * wgt;
    }
    float disp = swp / sw;

    // wave32 tree reduction, one atomic per block
#pragma unroll
    for (int off = 16; off > 0; off >>= 1)
      disp += __shfl_xor(disp, off, 32);
    if (lane == 0)
      unsafeAtomicAdd(out, disp);
  }
}

extern "C" void kernel_launch(void* const* d_in, const int* in_sizes, int n_in,
                              void* d_out, int out_size, void* d_ws, size_t ws_size,
                              hipStream_t stream) {
  const float* preds   = (const float*)d_in[0];
  const float* gts     = (const float*)d_in[1];
  const float* normals = (const float*)d_in[2];
  float* out = (float*)d_out;

  projloss_zero<<<1, 1, 0, stream>>>(out);
  projloss_kernel<<<dim3(Bb * (Mm / ROWS)), dim3(WAVES * 32), 0, stream>>>(
      preds, gts, normals, out);
}